// vit_mse_losses_16879221473428
// MI455X (gfx1250) — compile-verified
//
#include <hip/hip_runtime.h>

// MSE(input, patchify(target)) for input (64,64,12288) f32, target (64,3,512,512) f32.
// Memory-bound streaming reduction: float4 NT loads on both tensors (patchify folded
// into addressing), squared diffs accumulated through V_WMMA_F32_16X16X4_F32 with a
// ones B-matrix (C accumulates 16x each value -> scale by 1/16), deterministic
// two-stage tree reduction (no float atomics).

typedef __attribute__((ext_vector_type(2))) float v2f;
typedef __attribute__((ext_vector_type(4))) float f4v;
typedef __attribute__((ext_vector_type(8))) float v8f;

#define TOTAL_ELEMS 50331648u          // 64*64*12288
#define Q_TOTAL     12582912u          // TOTAL_ELEMS / 4  (float4 units)
#define BLOCKS      4096
#define THREADS     256                // 8 waves of 32

__global__ __launch_bounds__(THREADS) void mse_patchify_wmma_stage1(
    const float* __restrict__ in, const float* __restrict__ tgt,
    float* __restrict__ partials) {
  const unsigned tid = blockIdx.x * THREADS + threadIdx.x;
  const unsigned stride = BLOCKS * THREADS;          // 1,048,576 threads: 12 iters each

  v8f acc = {};                                      // 16x16 f32 WMMA accumulator
  v2f bones; bones[0] = 1.0f; bones[1] = 1.0f;       // B = ones (4x16)

  for (unsigned q = tid; q < Q_TOTAL; q += stride) { // uniform trip count: EXEC stays ~0
    const unsigned e = q << 2;                       // element index, < 2^26
    // fold patchify permutation into target address
    const unsigned r     = e & 4095u;                // py*64 + px
    const unsigned m     = e >> 12;                  // (n*64+patch)*3 + c
    const unsigned w     = m / 3u;
    const unsigned c     = m - w * 3u;
    const unsigned n     = w >> 6;
    const unsigned patch = w & 63u;
    const unsigned i     = patch >> 3, j = patch & 7u;
    const unsigned py    = r >> 6,     px = r & 63u;
    const size_t te = ((size_t)(n * 3u + c) << 18) +
                      ((size_t)((i << 6) + py) << 9) + (size_t)((j << 6) + px);

    f4v a = __builtin_nontemporal_load((const f4v*)(in  + (size_t)e));
    f4v b = __builtin_nontemporal_load((const f4v*)(tgt + te));

    f4v d;
    d.x = a.x - b.x; d.y = a.y - b.y; d.z = a.z - b.z; d.w = a.w - b.w;

    v2f a0; a0[0] = d.x * d.x; a0[1] = d.y * d.y;    // A matrix 16x4 f32 (2 VGPRs/lane)
    v2f a1; a1[0] = d.z * d.z; a1[1] = d.w * d.w;

    // D = A x ones + C : every A entry lands (replicated 16x over columns) in C
    acc = __builtin_amdgcn_wmma_f32_16x16x4_f32(false, a0, false, bones,
                                                (short)0, acc, false, false);
    acc = __builtin_amdgcn_wmma_f32_16x16x4_f32(false, a1, false, bones,
                                                (short)0, acc, false, false);
  }

  // wave-level sum of all 256 C entries = 16 * (sum of squared diffs this wave)
  float p = 0.0f;
  #pragma unroll
  for (int k = 0; k < 8; ++k) p += acc[k];
  p *= 0.0625f;                                      // undo 16x replication (exact)

  #pragma unroll
  for (int off = 16; off >= 1; off >>= 1)            // wave32 butterfly
    p += __shfl_xor(p, off, 32);

  __shared__ float smem[THREADS / 32];
  const unsigned lane = threadIdx.x & 31u;
  const unsigned wid  = threadIdx.x >> 5;
  if (lane == 0) smem[wid] = p;
  __syncthreads();
  if (threadIdx.x == 0) {
    float s = 0.0f;
    #pragma unroll
    for (int k = 0; k < THREADS / 32; ++k) s += smem[k];
    partials[blockIdx.x] = s;                        // deterministic per-block partial
  }
}

__global__ __launch_bounds__(THREADS) void mse_patchify_wmma_stage2(
    const float* __restrict__ partials, float* __restrict__ out) {
  float s = 0.0f;
  for (unsigned k = threadIdx.x; k < BLOCKS; k += THREADS)
    s += partials[k];
  #pragma unroll
  for (int off = 16; off >= 1; off >>= 1)
    s += __shfl_xor(s, off, 32);

  __shared__ float smem[THREADS / 32];
  const unsigned lane = threadIdx.x & 31u;
  const unsigned wid  = threadIdx.x >> 5;
  if (lane == 0) smem[wid] = s;
  __syncthreads();
  if (threadIdx.x == 0) {
    float t = 0.0f;
    #pragma unroll
    for (int k = 0; k < THREADS / 32; ++k) t += smem[k];
    out[0] = t * (1.0f / (float)TOTAL_ELEMS);        // mean
  }
}

extern "C" void kernel_launch(void* const* d_in, const int* in_sizes, int n_in,
                              void* d_out, int out_size, void* d_ws, size_t ws_size,
                              hipStream_t stream) {
  const float* input  = (const float*)d_in[0];   // (64, 64, 12288) f32
  const float* target = (const float*)d_in[1];   // (64, 3, 512, 512) f32
  float* out      = (float*)d_out;               // scalar f32
  float* partials = (float*)d_ws;                // BLOCKS floats of scratch

  mse_patchify_wmma_stage1<<<BLOCKS, THREADS, 0, stream>>>(input, target, partials);
  mse_patchify_wmma_stage2<<<1, THREADS, 0, stream>>>(partials, out);
}